// PaletteLoss_38757784879886
// MI455X (gfx1250) — compile-verified
//
#include <hip/hip_runtime.h>
#include <math.h>

typedef float v2f __attribute__((ext_vector_type(2)));
typedef float v8f __attribute__((ext_vector_type(8)));

#define HW               (256 * 256)
#define BATCH            16
#define KPAL             64
#define BLOCKS_PER_BATCH 32
#define NBLOCKS          (BATCH * BLOCKS_PER_BATCH)          // 512
#define TILES_PER_BLOCK  (HW / 16 / BLOCKS_PER_BATCH)        // 128 tiles of 16 pixels
#define TILES_PER_WAVE   (TILES_PER_BLOCK / 8)               // 16
#define ALPHA_C          0.001f

// ---------------------------------------------------------------------------
// Kernel 1: per-pixel  min_k ||x - p_k||^2 - ||x||^2  =  min_k(||p_k||^2 - 2 x.p_k)
// via V_WMMA_F32_16X16X4_F32 with C = inline 0.
//
// Trick: the K=3 slot of the 16x4 x 4x16 matmul carries the bias.
//   A (M=palette, K=channel): lanes 0-15  -> (K0,K1) = (-2r, -2g) of palette M=lane
//                             lanes 16-31 -> (K2,K3) = (-2b, ||p||^2)
//   B (K=channel, N=pixel):   lanes 0-15  -> (K0,K1) = (R, G) of pixel N=lane
//                             lanes 16-31 -> (K2,K3) = (B, 1.0)
//   => D[M,N] = ||p_M||^2 - 2 x_N . p_M   with C = 0 (no resident C registers,
//      no per-iteration register copies).
//
// D layout puts 8 palette rows in each lane's D octet (lower half: rows 0-7 of
// a group, upper half: rows 8-15), so the 64-way palette min is 31 VALU mins
// plus a single shfl_xor(16) to merge the wave halves.
// ---------------------------------------------------------------------------
__global__ __launch_bounds__(256) void pq_mse_kernel(
    const float* __restrict__ pal,    // [16,64,3]
    const float* __restrict__ img,    // [16,3,256,256]
    float* __restrict__ partial)      // [NBLOCKS]
{
    const int tid  = threadIdx.x;
    const int lane = tid & 31;
    const int wave = tid >> 5;
    const int b      = blockIdx.x / BLOCKS_PER_BATCH;
    const int blkInB = blockIdx.x % BLOCKS_PER_BATCH;
    const bool lo = (lane < 16);
    const int  lm = lane & 15;

    // ---- resident A operands (per block), one per palette group of 16 ----
    const float* pb = pal + (size_t)b * KPAL * 3;
    v2f amat[4];
#pragma unroll
    for (int g = 0; g < 4; ++g) {
        const int k = g * 16 + lm;
        const float pr = pb[k * 3 + 0];
        const float pg = pb[k * 3 + 1];
        const float pz = pb[k * 3 + 2];
        const float n2 = pr * pr + pg * pg + pz * pz;
        amat[g].x = -2.0f * (lo ? pr : pz);
        amat[g].y = lo ? (-2.0f * pg) : n2;       // K=3 slot carries ||p||^2
    }

    const size_t ibase = (size_t)b * 3 * HW;
    // Lane-uniform pointer select: lower half streams the R plane, upper half
    // the B plane; the G plane is loaded unconditionally by all lanes.
    const float* xptr = img + ibase + (lo ? 0 : 2 * HW);
    const float* gp   = img + ibase + HW;

    const v8f zeroC = {};                         // lowers to inline SRC2 = 0

    float accMin = 0.0f;   // sum over pixels of min_k score (replicated x2)
    float accN   = 0.0f;   // sum over pixels of ||x||^2 (split across lanes)

#pragma unroll 2
    for (int i = 0; i < TILES_PER_WAVE; ++i) {    // compile-time trip count
        const int t = wave + 8 * i;
        const int P = (blkInB * TILES_PER_BLOCK + t) * 16;   // tile base pixel

        const float xv = xptr[P + lm];            // R (lanes 0-15) / B (16-31)
        const float gv = gp[P + lm];              // G (value ignored on 16-31)
        const float yv = lo ? gv : 1.0f;          // K=3 slot carries 1.0

        // streaming sum of ||x||^2: r^2+g^2 on lower half, b^2 on upper half
        accN += xv * xv + (lo ? gv * gv : 0.0f);

        v2f bmx;
        bmx.x = xv;
        bmx.y = yv;

        // 4 WMMAs with inline-zero C; same B (pixels) reused across groups.
        v8f d0 = __builtin_amdgcn_wmma_f32_16x16x4_f32(
            false, amat[0], false, bmx, (short)0, zeroC, false, false);
        v8f d1 = __builtin_amdgcn_wmma_f32_16x16x4_f32(
            false, amat[1], false, bmx, (short)0, zeroC, false, false);
        v8f d2 = __builtin_amdgcn_wmma_f32_16x16x4_f32(
            false, amat[2], false, bmx, (short)0, zeroC, false, false);
        v8f d3 = __builtin_amdgcn_wmma_f32_16x16x4_f32(
            false, amat[3], false, bmx, (short)0, zeroC, false, false);

        // 64-way palette min, almost entirely in VALU (v_min3 chains):
        float m = fminf(fminf(d0[0], d1[0]), fminf(d2[0], d3[0]));
#pragma unroll
        for (int j = 1; j < 8; ++j) {
            const float mj = fminf(fminf(d0[j], d1[j]), fminf(d2[j], d3[j]));
            m = fminf(m, mj);
        }
        // merge the two wave halves (palette rows 0-7 vs 8-15): 1 shuffle
        m = fminf(m, __shfl_xor(m, 16, 32));
        accMin += m;                              // replicated x2 -> /2 later
    }

    // ---- deterministic wave + block reduction ----
    float v = accMin * 0.5f + accN;               // undo x2 replication
#pragma unroll
    for (int s = 1; s < 32; s <<= 1) v += __shfl_xor(v, s, 32);

    __shared__ float wsum[8];
    if (lane == 0) wsum[wave] = v;
    __syncthreads();
    if (tid == 0) {
        float s = 0.0f;
#pragma unroll
        for (int w = 0; w < 8; ++w) s += wsum[w];
        partial[blockIdx.x] = s;
    }
}

// ---------------------------------------------------------------------------
// Kernel 2 (single block): reduce block partials, compute the tiny pairwise
// palette term (sum over full KxK matrix = 2x the upper triangle), and emit
// the final scalar.  Fixed-order reductions -> deterministic.
// ---------------------------------------------------------------------------
__global__ __launch_bounds__(256) void pq_finish_kernel(
    const float* __restrict__ pal,
    const float* __restrict__ partial,
    float* __restrict__ out)
{
    const int tid = threadIdx.x;

    float s = 0.0f;
    for (int i = tid; i < NBLOCKS; i += 256) s += partial[i];

    float p = 0.0f;
    for (int idx = tid; idx < BATCH * KPAL * KPAL; idx += 256) {
        const int b   = idx >> 12;
        const int rem = idx & 4095;
        const int i   = rem >> 6;
        const int j   = rem & 63;
        const float* pb = pal + (size_t)b * KPAL * 3;
        const float dx = pb[i * 3 + 0] - pb[j * 3 + 0];
        const float dy = pb[i * 3 + 1] - pb[j * 3 + 1];
        const float dz = pb[i * 3 + 2] - pb[j * 3 + 2];
        p += sqrtf(dx * dx + dy * dy + dz * dz);
    }

    __shared__ float sm[256];
    __shared__ float pm[256];
    sm[tid] = s;
    pm[tid] = p;
    __syncthreads();
    for (int step = 128; step > 0; step >>= 1) {
        if (tid < step) {
            sm[tid] += sm[tid + step];
            pm[tid] += pm[tid + step];
        }
        __syncthreads();
    }

    if (tid == 0) {
        const float mse   = sm[0] / (float)(BATCH * 3 * HW);
        const float ploss = (0.5f * pm[0]) / ((KPAL * (KPAL - 1) / 2) * (float)BATCH);
        out[0] = mse - ALPHA_C * ploss;
    }
}

// ---------------------------------------------------------------------------
extern "C" void kernel_launch(void* const* d_in, const int* in_sizes, int n_in,
                              void* d_out, int out_size, void* d_ws, size_t ws_size,
                              hipStream_t stream)
{
    (void)in_sizes; (void)n_in; (void)out_size; (void)ws_size;
    const float* palettes = (const float*)d_in[0];   // [16,64,3]
    const float* images   = (const float*)d_in[1];   // [16,3,256,256]
    float* out            = (float*)d_out;           // scalar f32
    float* partial        = (float*)d_ws;            // NBLOCKS floats (fully rewritten each call)

    pq_mse_kernel<<<NBLOCKS, 256, 0, stream>>>(palettes, images, partial);
    pq_finish_kernel<<<1, 256, 0, stream>>>(palettes, partial, out);
}